// AttentionGuidedDynamicRangeDWConv3D_46359876993360
// MI455X (gfx1250) — compile-verified
//
#include <hip/hip_runtime.h>
#include <math.h>

#define B_N   4
#define C_CH  96
#define DDIM  16
#define HDIM  56
#define WDIM  56
#define GDIM  96
#define HID   24
#define NB    3
#define NCOL  (C_CH + GDIM)          // 192
#define SPATIAL (DDIM * HDIM * WDIM) // 50176
#define LN_EPS 1e-5f

typedef float v2f __attribute__((ext_vector_type(2)));
typedef float v4f __attribute__((ext_vector_type(4)));
typedef float v8f __attribute__((ext_vector_type(8)));

// ---------------------------------------------------------------------------
// Kernel 1: global mean pool  feat[b,c] = mean(x[b,c,:,:,:])
// One block per (b,c); 256 threads; B128 loads; LDS tree reduction.
// Also warms L2 with x (77.1 MB fully fits in the 192 MB L2).
// ---------------------------------------------------------------------------
__global__ void feat_reduce_kernel(const float* __restrict__ x,
                                   float* __restrict__ feat) {
    const int bc = blockIdx.x;                         // 0 .. B*C-1
    const v4f* p = (const v4f*)(x + (size_t)bc * SPATIAL);
    float s = 0.f;
    for (int i = threadIdx.x; i < SPATIAL / 4; i += 256) {
        v4f v = p[i];
        s += v.x + v.y + v.z + v.w;
    }
    __shared__ float red[256];
    red[threadIdx.x] = s;
    __syncthreads();
    for (int off = 128; off > 0; off >>= 1) {
        if (threadIdx.x < off) red[threadIdx.x] += red[threadIdx.x + off];
        __syncthreads();
    }
    if (threadIdx.x == 0) feat[bc] = red[0] * (1.0f / (float)SPATIAL);
}

// ---------------------------------------------------------------------------
// Kernel 2: gating MLP on a single wave32.
//   g = [feat | guidance] (4x192) -> LayerNorm -> WMMA GEMM (192->24)
//   -> erf GELU -> scalar 24x3 -> softmax -> wts[4][3]
// GEMM = two 16x16 output tiles, 48 chained V_WMMA_F32_16X16X4_F32 each.
// A (16x192) and B^T (32x192) are zero-padded in LDS so every fragment load
// is an unconditional ds_load_b64 — no EXEC predication anywhere near WMMA.
// ---------------------------------------------------------------------------
__global__ void gate_kernel(const float* __restrict__ feat,
                            const float* __restrict__ guidance,
                            const float* __restrict__ ln_g,
                            const float* __restrict__ ln_b,
                            const float* __restrict__ w1p,
                            const float* __restrict__ b1p,
                            const float* __restrict__ w2p,
                            const float* __restrict__ b2p,
                            float* __restrict__ wts) {
    __shared__ float gn[16][NCOL];       // A: rows 0..3 data, 4..15 zero
    __shared__ float w1t[32][NCOL];      // B^T: rows 0..23 = w1 cols, 24..31 zero
    __shared__ float mu[B_N];
    __shared__ float rs[B_N];
    __shared__ float hbuf[B_N][HID];
    __shared__ float logits[B_N][NB];

    const int lane = threadIdx.x;        // 0..31, one wave

    // zero-pad A rows 4..15
    for (int idx = lane; idx < 12 * NCOL; idx += 32)
        gn[4 + idx / NCOL][idx % NCOL] = 0.f;
    // stage w1 transposed (+ zero pad cols 24..31)
    for (int idx = lane; idx < 32 * NCOL; idx += 32) {
        const int n = idx / NCOL, k = idx % NCOL;
        w1t[n][k] = (n < HID) ? w1p[k * HID + n] : 0.f;
    }
    // gather raw g into rows 0..3
    for (int idx = lane; idx < B_N * NCOL; idx += 32) {
        const int r = idx / NCOL, col = idx % NCOL;
        gn[r][col] = (col < C_CH) ? feat[r * C_CH + col]
                                  : guidance[r * GDIM + (col - C_CH)];
    }
    __syncthreads();

    // per-row LayerNorm statistics (4 rows, cheap serial scan)
    if (lane < B_N) {
        float m = 0.f;
        for (int j = 0; j < NCOL; ++j) m += gn[lane][j];
        m *= (1.0f / NCOL);
        float v = 0.f;
        for (int j = 0; j < NCOL; ++j) { float d = gn[lane][j] - m; v += d * d; }
        v *= (1.0f / NCOL);
        mu[lane] = m;
        rs[lane] = rsqrtf(v + LN_EPS);
    }
    __syncthreads();

    // normalize + affine (rows 0..3 only)
    for (int idx = lane; idx < B_N * NCOL; idx += 32) {
        const int r = idx / NCOL, col = idx % NCOL;
        gn[r][col] = (gn[r][col] - mu[r]) * rs[r] * ln_g[col] + ln_b[col];
    }
    __syncthreads();

    // ---- WMMA GEMM ----
    // A 16x4 f32: lanes 0-15 -> M=lane, K={0,1}; lanes 16-31 -> M=lane-16, K={2,3}
    // B 4x16 f32: VGPR0 = rows K=0/K=2 per lane half, VGPR1 = K=1/K=3
    const int half = lane >> 4;
    const int l    = lane & 15;
    const float* arow  = &gn[l][0];
    const float* b0row = &w1t[l][0];
    const float* b1row = &w1t[16 + l][0];
    v8f c0 = {};                          // tile N = 0..15
    v8f c1 = {};                          // tile N = 16..31 (16..23 valid)
    for (int kk = 0; kk < NCOL / 4; ++kk) {
        const int kb = kk * 4 + half * 2;         // even -> 8B aligned
        const v2f a   = *(const v2f*)(arow + kb);
        const v2f b0  = *(const v2f*)(b0row + kb);
        const v2f b1v = *(const v2f*)(b1row + kb);
        c0 = __builtin_amdgcn_wmma_f32_16x16x4_f32(false, a, false, b0,
                                                   (short)0, c0, false, false);
        c1 = __builtin_amdgcn_wmma_f32_16x16x4_f32(false, a, false, b1v,
                                                   (short)0, c1, false, false);
    }

    // unpack C tiles, add bias, exact (erf) GELU, stash to LDS
    #pragma unroll
    for (int r = 0; r < 8; ++r) {
        const int M = half ? (r + 8) : r;
        if (M < B_N) {
            float v0 = c0[r] + b1p[l];
            v0 = 0.5f * v0 * (1.0f + erff(v0 * 0.70710678118654752f));
            hbuf[M][l] = v0;
            const int n1 = 16 + l;
            if (n1 < HID) {
                float v1 = c1[r] + b1p[n1];
                v1 = 0.5f * v1 * (1.0f + erff(v1 * 0.70710678118654752f));
                hbuf[M][n1] = v1;
            }
        }
    }
    __syncthreads();

    // tiny second layer: logits[b][j] = b2[j] + h[b,:] . w2[:,j]
    if (lane < B_N * NB) {
        const int b = lane / NB, j = lane % NB;
        float s = b2p[j];
        for (int k = 0; k < HID; ++k) s += hbuf[b][k] * w2p[k * NB + j];
        logits[b][j] = s;
    }
    __syncthreads();

    // softmax over branches
    if (lane < B_N) {
        float m = fmaxf(logits[lane][0], fmaxf(logits[lane][1], logits[lane][2]));
        float e0 = __expf(logits[lane][0] - m);
        float e1 = __expf(logits[lane][1] - m);
        float e2 = __expf(logits[lane][2] - m);
        float inv = 1.0f / (e0 + e1 + e2);
        wts[lane * NB + 0] = e0 * inv;
        wts[lane * NB + 1] = e1 * inv;
        wts[lane * NB + 2] = e2 * inv;
    }
}

// ---------------------------------------------------------------------------
// Kernel 3: fold softmax weights into per-(b,c) conv weights and bias.
//   wscaled[b][c][i][t] = wts[b][i] * convw[i][c][t]      (t = 27 taps)
//   biaseff[b][c]       = sum_i wts[b][i] * convb[i][c]
// ---------------------------------------------------------------------------
__global__ void fold_kernel(const float* __restrict__ convw,
                            const float* __restrict__ convb,
                            const float* __restrict__ wts,
                            float* __restrict__ wscaled,
                            float* __restrict__ biaseff) {
    const int idx = blockIdx.x * 256 + threadIdx.x;
    const int total = B_N * C_CH * NB * 27;
    if (idx < total) {
        const int t = idx % 27;
        int tmp = idx / 27;
        const int i = tmp % NB;
        tmp /= NB;
        const int c = tmp % C_CH;
        const int b = tmp / C_CH;
        wscaled[idx] = wts[b * NB + i] * convw[(i * C_CH + c) * 27 + t];
    }
    if (idx < B_N * C_CH) {
        const int c = idx % C_CH, b = idx / C_CH;
        float s = 0.f;
        for (int i = 0; i < NB; ++i) s += wts[b * NB + i] * convb[i * C_CH + c];
        biaseff[idx] = s;
    }
}

// ---------------------------------------------------------------------------
// Kernel 4: fused 3-branch dilated depthwise 3x3x3 conv.
// One block per (b, c, z) plane (56x56 outputs). x is L2-resident, so direct
// global loads; 81 alpha-scaled weights + effective bias in LDS. Interior
// voxels (>=3 from every face) take a branch-free 81-FMA fast path.
// ---------------------------------------------------------------------------
__global__ void dwconv_kernel(const float* __restrict__ x,
                              const float* __restrict__ wscaled,
                              const float* __restrict__ biaseff,
                              float* __restrict__ out) {
    const int bcd = blockIdx.x;
    const int z  = bcd % DDIM;
    const int bc = bcd / DDIM;

    __shared__ float wsm[NB * 27];
    __shared__ float biass;
    if (threadIdx.x < NB * 27) wsm[threadIdx.x] = wscaled[bc * NB * 27 + threadIdx.x];
    if (threadIdx.x == NB * 27) biass = biaseff[bc];
    __syncthreads();

    const float* xp = x + (size_t)bc * SPATIAL;
    float* op = out + (size_t)bc * SPATIAL + (size_t)z * (HDIM * WDIM);
    const bool zint = (z >= 3) && (z <= DDIM - 4);   // block-uniform

    for (int hw = threadIdx.x; hw < HDIM * WDIM; hw += 256) {
        const int y  = hw / WDIM;
        const int xw = hw % WDIM;
        float acc = biass;
        if (zint && (y >= 3) && (y <= HDIM - 4) && (xw >= 3) && (xw <= WDIM - 4)) {
            // interior: no bounds checks, straight-line 81 FMAs
            const float* base = xp + ((size_t)z * HDIM + y) * WDIM + xw;
            #pragma unroll
            for (int i = 0; i < NB; ++i) {
                const int di = i + 1;
                const float* wp = &wsm[i * 27];
                #pragma unroll
                for (int kz = 0; kz < 3; ++kz) {
                    #pragma unroll
                    for (int ky = 0; ky < 3; ++ky) {
                        const float* row = base + ((kz - 1) * di * HDIM
                                                 + (ky - 1) * di) * WDIM;
                        #pragma unroll
                        for (int kx = 0; kx < 3; ++kx)
                            acc = fmaf(wp[(kz * 3 + ky) * 3 + kx],
                                       row[(kx - 1) * di], acc);
                    }
                }
            }
        } else {
            // boundary: per-axis unsigned-compare predication
            #pragma unroll
            for (int i = 0; i < NB; ++i) {
                const int di = i + 1;
                const float* wp = &wsm[i * 27];
                #pragma unroll
                for (int kz = 0; kz < 3; ++kz) {
                    const int z2 = z + (kz - 1) * di;
                    if ((unsigned)z2 >= (unsigned)DDIM) continue;
                    #pragma unroll
                    for (int ky = 0; ky < 3; ++ky) {
                        const int y2 = y + (ky - 1) * di;
                        if ((unsigned)y2 >= (unsigned)HDIM) continue;
                        const float* row = xp + ((size_t)z2 * HDIM + y2) * WDIM;
                        #pragma unroll
                        for (int kx = 0; kx < 3; ++kx) {
                            const int x2 = xw + (kx - 1) * di;
                            if ((unsigned)x2 >= (unsigned)WDIM) continue;
                            acc = fmaf(wp[(kz * 3 + ky) * 3 + kx], row[x2], acc);
                        }
                    }
                }
            }
        }
        op[hw] = acc;
    }
}

// ---------------------------------------------------------------------------
extern "C" void kernel_launch(void* const* d_in, const int* in_sizes, int n_in,
                              void* d_out, int out_size, void* d_ws, size_t ws_size,
                              hipStream_t stream) {
    const float* x        = (const float*)d_in[0];
    const float* guidance = (const float*)d_in[1];
    const float* convw    = (const float*)d_in[2];
    const float* convb    = (const float*)d_in[3];
    const float* ln_g     = (const float*)d_in[4];
    const float* ln_b     = (const float*)d_in[5];
    const float* w1       = (const float*)d_in[6];
    const float* b1       = (const float*)d_in[7];
    const float* w2       = (const float*)d_in[8];
    const float* b2       = (const float*)d_in[9];

    float* ws      = (float*)d_ws;
    float* feat    = ws;            // 384 floats
    float* wts     = ws + 384;      // 12 floats
    float* biaseff = ws + 512;      // 384 floats
    float* wscaled = ws + 1024;     // 31104 floats

    feat_reduce_kernel<<<B_N * C_CH, 256, 0, stream>>>(x, feat);
    gate_kernel<<<1, 32, 0, stream>>>(feat, guidance, ln_g, ln_b,
                                      w1, b1, w2, b2, wts);
    const int foldN = B_N * C_CH * NB * 27;
    fold_kernel<<<(foldN + 255) / 256, 256, 0, stream>>>(convw, convb, wts,
                                                         wscaled, biaseff);
    dwconv_kernel<<<B_N * C_CH * DDIM, 256, 0, stream>>>(x, wscaled, biaseff,
                                                         (float*)d_out);
}